// SpectralCF_71657234366494
// MI455X (gfx1250) — compile-verified
//
#include <hip/hip_runtime.h>
#include <math.h>

#define D 256
#define LAYERS 3
#define SLOTS (LAYERS + 1)
#define LDSPITCH 260   // padded row stride in floats (260 % 64 = 4 -> conflict-free A-frag reads)

typedef float v2f __attribute__((ext_vector_type(2)));
typedef float v8f __attribute__((ext_vector_type(8)));

// ---------------------------------------------------------------------------
// Init: stacked[n, 0, :] = embeddings[n, :]; stacked[n, 1..3, :] = 0
// stacked layout: float element (n*SLOTS + l)*D + d  (== reference [N, L+1, D])
// ---------------------------------------------------------------------------
__global__ void init_stacked(const float4* __restrict__ emb,
                             float4* __restrict__ stacked, long total4) {
    long i = (long)blockIdx.x * blockDim.x + threadIdx.x;
    if (i >= total4) return;
    int c4 = (int)(i & 63);          // float4 column within row (D/4 = 64)
    int l  = (int)((i >> 6) & 3);    // slot
    long n = i >> 8;                 // node
    float4 v = make_float4(0.f, 0.f, 0.f, 0.f);
    if (l == 0) v = emb[n * (D / 4) + c4];
    stacked[i] = v;
}

// ---------------------------------------------------------------------------
// SpMM scatter: one wave32 per edge.
//   y[row] += val * x[col]   (x = slot `layer`, y = slot `layer+1`)
// Lanes do two coalesced float4 gathers (L2-resident: x slab is 102.4 MB,
// fits the 192 MB L2) and 8 f32 HW atomics (no-return -> STOREcnt path).
// ---------------------------------------------------------------------------
__global__ void spmm_scatter(float* __restrict__ stacked,
                             const int* __restrict__ rows,
                             const int* __restrict__ cols,
                             const float* __restrict__ vals,
                             int layer, int n_edges) {
    long gid = (long)blockIdx.x * blockDim.x + threadIdx.x;
    int e    = (int)(gid >> 5);
    int lane = (int)(gid & 31);
    if (e >= n_edges) return;

    int   r = rows[e];
    int   c = cols[e];
    float v = vals[e];

    const float4* src = (const float4*)(stacked + ((long)c * SLOTS + layer) * D);
    float*        dst = stacked + ((long)r * SLOTS + layer + 1) * D;

    float4 x0 = src[lane];        // bytes [0, 512)   fully coalesced across wave
    float4 x1 = src[32 + lane];   // bytes [512,1024)

    int o0 = 4 * lane;
    unsafeAtomicAdd(dst + o0 + 0, v * x0.x);
    unsafeAtomicAdd(dst + o0 + 1, v * x0.y);
    unsafeAtomicAdd(dst + o0 + 2, v * x0.z);
    unsafeAtomicAdd(dst + o0 + 3, v * x0.w);
    int o1 = 128 + 4 * lane;
    unsafeAtomicAdd(dst + o1 + 0, v * x1.x);
    unsafeAtomicAdd(dst + o1 + 1, v * x1.y);
    unsafeAtomicAdd(dst + o1 + 2, v * x1.z);
    unsafeAtomicAdd(dst + o1 + 3, v * x1.w);
}

// Fast sigmoid: hardware exp (exp2-based) + hardware reciprocal (v_rcp_f32,
// ~1 ulp) -- avoids the correctly-rounded v_div_scale/fixup sequence.
__device__ __forceinline__ float fast_sigmoid(float x) {
    return __builtin_amdgcn_rcpf(1.0f + __expf(-x));
}

// ---------------------------------------------------------------------------
// Dense GEMM + sigmoid, in place on slot layer+1:
//   x <- sigmoid(y @ W),  y = slot layer+1 (post-scatter), W = filters[layer]
// Block = 128 threads (4 waves), handles 16 rows x 256 cols.
// Each wave: 4 N-tiles of 16, 64 steps of V_WMMA_F32_16X16X4_F32 each
// (exact fp32 matmul on the matrix pipe; GEMM is ~13 GFLOP/layer, noise
// next to the L2-bound SpMM, so fp32 exactness > low-precision throughput).
// ---------------------------------------------------------------------------
__global__ __launch_bounds__(128) void gemm_sigmoid(float* __restrict__ stacked,
                                                    const float* __restrict__ W,
                                                    int layer, int n_nodes) {
    __shared__ float ytile[16 * LDSPITCH];

    const int m0  = blockIdx.x * 16;
    const int tid = threadIdx.x;

    // Stage 16x256 A-tile into LDS (coalesced float4 loads).
    for (int i = 0; i < 8; ++i) {
        int f   = tid + i * 128;      // 1024 float4s total
        int rr  = f >> 6;             // row 0..15
        int cc4 = f & 63;             // float4 col 0..63
        long n  = m0 + rr;
        if (n >= n_nodes) n = n_nodes - 1;   // clamp (N=100000 is exact anyway)
        float4 v = *(const float4*)(stacked + (n * SLOTS + layer + 1) * D + cc4 * 4);
        *(float4*)(&ytile[rr * LDSPITCH + cc4 * 4]) = v;
    }
    __syncthreads();

    const int  lane     = tid & 31;
    const int  wave     = tid >> 5;
    const int  g        = lane >> 4;   // lane group: 0 -> K=0,1 ; 1 -> K=2,3
    const int  lm       = lane & 15;   // M (for A) / N (for B) within tile
    const bool fulltile = (m0 + 16 <= n_nodes);  // uniform: branch-free hot path

    for (int t = 0; t < 4; ++t) {
        const int n0 = (wave * 4 + t) * 16;
        v8f acc = {0.f, 0.f, 0.f, 0.f, 0.f, 0.f, 0.f, 0.f};

        for (int s = 0; s < 64; ++s) {
            const int k0 = 4 * s;
            v2f a, b;
            // A 16x4 f32 fragment from LDS (ISA 7.12.2 layout)
            a[0] = ytile[lm * LDSPITCH + k0 + 2 * g];
            a[1] = ytile[lm * LDSPITCH + k0 + 2 * g + 1];
            // B 4x16 f32 fragment from W (row-major [K=256, N=256], L2-resident)
            b[0] = W[(k0 + 2 * g) * D + n0 + lm];
            b[1] = W[(k0 + 2 * g + 1) * D + n0 + lm];
            acc = __builtin_amdgcn_wmma_f32_16x16x4_f32(
                false, a, false, b, (short)0, acc, false, false);
        }

        // D layout: VGPR r -> row m0 + r + 8*g, col n0 + lm. Sigmoid + in-place
        // store. Base pointer + r*(SLOTS*D) keeps offsets in the 24-bit ioffset.
        float* outbase =
            stacked + ((long)(m0 + 8 * g) * SLOTS + layer + 1) * D + n0 + lm;
        if (fulltile) {
            #pragma unroll
            for (int r = 0; r < 8; ++r) {
                outbase[(long)r * (SLOTS * D)] = fast_sigmoid(acc[r]);
            }
        } else {
            #pragma unroll
            for (int r = 0; r < 8; ++r) {
                if (m0 + r + 8 * g < n_nodes) {
                    outbase[(long)r * (SLOTS * D)] = fast_sigmoid(acc[r]);
                }
            }
        }
    }
}

// ---------------------------------------------------------------------------
// Readout: mean over the 4 slots -> d_out[0 : N*D]
// ---------------------------------------------------------------------------
__global__ void mean_kernel(const float4* __restrict__ stacked,
                            float4* __restrict__ out, long total4) {
    long i = (long)blockIdx.x * blockDim.x + threadIdx.x;
    if (i >= total4) return;
    long n = i >> 6;
    int c4 = (int)(i & 63);
    long base = n * SLOTS * (D / 4) + c4;
    float4 s0 = stacked[base];
    float4 s1 = stacked[base + 64];
    float4 s2 = stacked[base + 128];
    float4 s3 = stacked[base + 192];
    float4 m;
    m.x = (s0.x + s1.x + s2.x + s3.x) * 0.25f;
    m.y = (s0.y + s1.y + s2.y + s3.y) * 0.25f;
    m.z = (s0.z + s1.z + s2.z + s3.z) * 0.25f;
    m.w = (s0.w + s1.w + s2.w + s3.w) * 0.25f;
    out[i] = m;
}

extern "C" void kernel_launch(void* const* d_in, const int* in_sizes, int n_in,
                              void* d_out, int out_size, void* d_ws, size_t ws_size,
                              hipStream_t stream) {
    const float* emb     = (const float*)d_in[0];
    const int*   rows    = (const int*)d_in[1];
    const int*   cols    = (const int*)d_in[2];
    const float* vals    = (const float*)d_in[3];
    const float* filters = (const float*)d_in[4];

    const int n_nodes = in_sizes[0] / D;
    const int n_edges = in_sizes[1];

    float* mean_out = (float*)d_out;
    float* stacked  = mean_out + (long)n_nodes * D;   // [N, L+1, D] region of d_out

    // 1) init stacked: slot0 = embeddings, slots 1..3 = 0
    long init4 = (long)n_nodes * SLOTS * (D / 4);
    init_stacked<<<(unsigned)((init4 + 255) / 256), 256, 0, stream>>>(
        (const float4*)emb, (float4*)stacked, init4);

    // 2) layers
    const int mtiles = (n_nodes + 15) / 16;
    for (int k = 0; k < LAYERS; ++k) {
        long th = (long)n_edges * 32;   // one wave32 per edge
        spmm_scatter<<<(unsigned)((th + 255) / 256), 256, 0, stream>>>(
            stacked, rows, cols, vals, k, n_edges);
        gemm_sigmoid<<<mtiles, 128, 0, stream>>>(
            stacked, filters + (long)k * D * D, k, n_nodes);
    }

    // 3) mean readout
    long mean4 = (long)n_nodes * (D / 4);
    mean_kernel<<<(unsigned)((mean4 + 255) / 256), 256, 0, stream>>>(
        (const float4*)stacked, (float4*)mean_out, mean4);
}